// WALContext_90683939488284
// MI455X (gfx1250) — compile-verified
//
#include <hip/hip_runtime.h>
#include <hip/hip_bf16.h>
#include <math.h>

// Problem dims (fixed by the reference)
constexpr int Bb = 8;     // batch
constexpr int Nn = 1000;  // points
constexpr int Cc = 4000;  // customers
constexpr int Dd = 128;   // embed dim

constexpr float LAMBDA_G = 1.938f;
constexpr float LAMBDA_F = 1.912f;
constexpr float TAU      = 3.5f;
// advantage = revenue*(0.19 - 0.075 - 0.005) - delivery - fixed
constexpr float NETM     = 0.19f - 0.075f - 0.005f;
constexpr float EPSV     = 1e-8f;

typedef __attribute__((ext_vector_type(2))) float v2f;
typedef __attribute__((ext_vector_type(8))) float v8f;

// ---------------------------------------------------------------------------
// Kernel 1: reward_improve[b,n].
// One wave32 per point. chosen_count selects which (if any) C-row to stream:
//   chosen==0 -> regular candidate (u_lj_reg, LAMBDA_G, reg distances)
//   chosen==1 -> food candidate    (u_lj_food, LAMBDA_F, food distances)
//   else      -> reward = 0 (all masks zero)
// Streaming reduction over C=4000 with float4 loads (128B/lane-iter, coalesced).
// u0/pop (32 KB per batch) are re-read per point but stay L2/L0-resident.
// ---------------------------------------------------------------------------
__global__ __launch_bounds__(256)
void reward_kernel(const int* __restrict__ chosen,
                   const float* __restrict__ u_reg,
                   const float* __restrict__ u_food,
                   const float* __restrict__ u0,
                   const float* __restrict__ pop,
                   const float* __restrict__ dist_reg,
                   const float* __restrict__ dist_food,
                   const float* __restrict__ fcp,
                   float* __restrict__ reward)
{
    const int wave = threadIdx.x >> 5;
    const int lane = threadIdx.x & 31;
    const int gidx = blockIdx.x * 8 + wave;       // 0 .. B*N-1  (grid = B*N/8)
    const int b = gidx / Nn;
    const int n = gidx - b * Nn;

    const int cc = chosen[b * Nn + n];
    const bool is_reg  = (cc < 1);
    const bool is_food = (cc >= 1) && (cc < 2);
    const bool active  = is_reg || is_food;

    float acc = 0.0f;
    if (active) {
        const float* urow = is_reg ? (u_reg  + ((size_t)(b * Nn + n)) * Cc)
                                   : (u_food + ((size_t)(b * Nn + n)) * Cc);
        const float4* u4  = (const float4*)urow;
        const float4* z4  = (const float4*)(u0  + (size_t)b * Cc);
        const float4* p4  = (const float4*)(pop + (size_t)b * Cc);
        // C/4 = 1000 float4 chunks per row
        for (int i = lane; i < Cc / 4; i += 32) {
            float4 u = u4[i];
            float4 z = z4[i];
            float4 p = p4[i];
            acc += p.x * (u.x / (z.x + u.x + EPSV));
            acc += p.y * (u.y / (z.y + u.y + EPSV));
            acc += p.z * (u.z / (z.z + u.z + EPSV));
            acc += p.w * (u.w / (z.w + u.w + EPSV));
        }
    }
    // wave32 shuffle reduction
    for (int m = 16; m > 0; m >>= 1) acc += __shfl_xor(acc, m, 32);

    if (lane == 0) {
        float rw = 0.0f;
        if (active) {
            const float lambda  = is_reg ? LAMBDA_G : LAMBDA_F;
            const float dist    = is_reg ? dist_reg[b * Nn + n] : dist_food[b * Nn + n];
            const float revenue = lambda * acc;
            const float delivery = TAU * dist * (1.0f / 1000.0f);
            const float fixedc   = fcp[b * Nn + n];   // fixed_cost_point is [B,1,N]
            const float adv = NETM * revenue - delivery - fixedc;
            rw = fmaxf(adv, 0.0f) * 0.01f;
        }
        reward[b * Nn + n] = rw;
    }
}

// ---------------------------------------------------------------------------
// Kernel 2: softmax over N per batch + context[b,d] = sum_n emb[b,n,d]*p[n].
// One 256-thread block per batch.
// ---------------------------------------------------------------------------
__global__ __launch_bounds__(256)
void softmax_context_kernel(const float* __restrict__ emb,
                            const float* __restrict__ reward,
                            float* __restrict__ ctx)
{
    __shared__ float sp[Nn];       // softmax numerators
    __shared__ float sred[8];      // per-wave partials
    __shared__ float sctx[2 * Dd]; // two n-half partial contexts

    const int b    = blockIdx.x;
    const int t    = threadIdx.x;
    const int lane = t & 31;
    const int wv   = t >> 5;
    const float* r = reward + b * Nn;

    // ---- max over N ----
    float m = -INFINITY;
    for (int n = t; n < Nn; n += 256) m = fmaxf(m, r[n]);
    for (int o = 16; o > 0; o >>= 1) m = fmaxf(m, __shfl_xor(m, o, 32));
    if (lane == 0) sred[wv] = m;
    __syncthreads();
    float bm = sred[0];
    for (int w = 1; w < 8; ++w) bm = fmaxf(bm, sred[w]);
    __syncthreads();

    // ---- exp + sum ----
    float s = 0.0f;
    for (int n = t; n < Nn; n += 256) {
        float e = __expf(r[n] - bm);
        sp[n] = e;
        s += e;
    }
    for (int o = 16; o > 0; o >>= 1) s += __shfl_xor(s, o, 32);
    if (lane == 0) sred[wv] = s;
    __syncthreads();
    float Z = 0.0f;
    for (int w = 0; w < 8; ++w) Z += sred[w];
    const float inv = 1.0f / Z;
    __syncthreads();

    // ---- context: d = t&127, two halves of n span handled by t>>7 ----
    const int d    = t & (Dd - 1);
    const int part = t >> 7;               // 0 or 1
    const int n0   = part * (Nn / 2);
    const int n1   = n0 + (Nn / 2);
    float acc = 0.0f;
    const float* eb = emb + ((size_t)b * Nn) * Dd;
    for (int n = n0; n < n1; ++n)
        acc += eb[(size_t)n * Dd + d] * sp[n];
    sctx[part * Dd + d] = acc;
    __syncthreads();
    if (t < Dd) ctx[b * Dd + t] = (sctx[t] + sctx[Dd + t]) * inv;
}

// ---------------------------------------------------------------------------
// Kernel 3: out[b,d] = sum_k ctx[b,k] * W[d,k] + bias[d]  via
// V_WMMA_F32_16X16X4_F32.  One wave per 16-wide d-tile (grid = D/16 = 8).
// A (16x4): M=lane%16 (batch rows, 8..15 zero-padded), VGPR0/1 hold
//   K = k0+{0,1} for lanes 0-15 and K = k0+{2,3} for lanes 16-31 (ISA layout).
// B (4x16): N=lane%16 = output dim in tile, same K split across lane halves.
// C/D (16x16 f32): VGPR r, lanes 0-15 -> row M=r  => batches 0..7 live there.
// Branchless fragment build keeps EXEC all-ones around the WMMA.
// ---------------------------------------------------------------------------
__global__ __launch_bounds__(32)
void out_gemm_wmma(const float* __restrict__ ctx,
                   const float* __restrict__ W,
                   const float* __restrict__ bias,
                   float* __restrict__ out)
{
    const int tile  = blockIdx.x;          // d-tile: columns tile*16 .. +15
    const int lane  = threadIdx.x;         // 0..31
    const int mrow  = lane & 15;           // A row (batch), C col (N)
    const int ksel  = (lane >> 4) * 2;     // 0 for lanes 0-15, 2 for 16-31
    const int ncol  = tile * 16 + mrow;    // global output dim for B frag
    const float msk = (mrow < 8) ? 1.0f : 0.0f;
    const int crow  = (mrow & 7);          // clamped in-bounds ctx row

    v8f c = {0.f, 0.f, 0.f, 0.f, 0.f, 0.f, 0.f, 0.f};
    for (int k0 = 0; k0 < Dd; k0 += 4) {
        v2f a, bmat;
        a.x = msk * ctx[crow * Dd + k0 + ksel + 0];
        a.y = msk * ctx[crow * Dd + k0 + ksel + 1];
        bmat.x = W[ncol * Dd + k0 + ksel + 0];
        bmat.y = W[ncol * Dd + k0 + ksel + 1];
        c = __builtin_amdgcn_wmma_f32_16x16x4_f32(
                /*neg_a=*/false, a, /*neg_b=*/false, bmat,
                /*c_mod=*/(short)0, c, /*reuse_a=*/false, /*reuse_b=*/false);
    }
    if (lane < 16) {
        const float bv = bias[tile * 16 + lane];
        #pragma unroll
        for (int rrow = 0; rrow < 8; ++rrow)          // batches 0..7
            out[rrow * Dd + tile * 16 + lane] = c[rrow] + bv;
    }
}

// ---------------------------------------------------------------------------
extern "C" void kernel_launch(void* const* d_in, const int* in_sizes, int n_in,
                              void* d_out, int out_size, void* d_ws, size_t ws_size,
                              hipStream_t stream)
{
    const float* emb       = (const float*)d_in[0];   // [B,N,D]
    const int*   chosen    = (const int*)  d_in[1];   // [B,N]
    const float* u_reg     = (const float*)d_in[2];   // [B,N,C]
    const float* u_food    = (const float*)d_in[3];   // [B,N,C]
    const float* u0        = (const float*)d_in[4];   // [B,C]
    const float* pop       = (const float*)d_in[5];   // [B,C,1]
    const float* dist_reg  = (const float*)d_in[6];   // [B,N,1]
    const float* dist_food = (const float*)d_in[7];   // [B,N,1]
    const float* fcp       = (const float*)d_in[8];   // [B,1,N]
    const float* W         = (const float*)d_in[9];   // [D,D]
    const float* bias      = (const float*)d_in[10];  // [D]
    float*       out       = (float*)d_out;           // [B,D]

    float* reward = (float*)d_ws;                     // B*N floats
    float* ctx    = reward + Bb * Nn;                 // B*D floats

    reward_kernel<<<(Bb * Nn) / 8, 256, 0, stream>>>(
        chosen, u_reg, u_food, u0, pop, dist_reg, dist_food, fcp, reward);

    softmax_context_kernel<<<Bb, 256, 0, stream>>>(emb, reward, ctx);

    out_gemm_wmma<<<Dd / 16, 32, 0, stream>>>(ctx, W, bias, out);
}